// MAMBA_CHUNK_STATE_kernel_32306744001007
// MI455X (gfx1250) — compile-verified
//
#include <hip/hip_runtime.h>
#include <hip/hip_bf16.h>

// Problem constants (from the reference)
#define BATCH   4
#define SEQLEN  8192
#define NGROUPS 1
#define DSTATE  128
#define NHEADS  32
#define HEADDIM 64
#define CHUNK   256
#define NCHUNKS 32

typedef float v2f __attribute__((ext_vector_type(2)));
typedef float v4f __attribute__((ext_vector_type(4)));
typedef float v8f __attribute__((ext_vector_type(8)));
typedef unsigned int u32x4 __attribute__((ext_vector_type(4)));
typedef int i32x4 __attribute__((ext_vector_type(4)));
typedef int i32x8 __attribute__((ext_vector_type(8)));

// LDS tile strides (floats). Padding chosen so the stride-row offset between
// lane halves (2*stride) is ≡ 16 (mod 64) banks -> conflict-free frag loads.
#define LHALF     128
#define SA_STRIDE 72     // 64 cols + 8 pad   (2*72  = 144, 144%64 = 16)
#define SB_STRIDE 136    // 128 cols + 8 pad  (2*136 = 272, 272%64 = 16)

__global__ __launch_bounds__(256)
void mamba_chunk_state_wmma(const float* __restrict__ Bmat,
                            const float* __restrict__ x,
                            const float* __restrict__ dt,
                            const float* __restrict__ dA,
                            float* __restrict__ out)
{
    __shared__ float sA[LHALF * SA_STRIDE];   // x * scale, [l][p]
    __shared__ float sB[LHALF * SB_STRIDE];   // B (TDM-filled), [l][n]
    __shared__ float sScale[CHUNK];

    const int tid = threadIdx.x;
    const int bid = blockIdx.x;                 // ((b*NCHUNKS + c)*NHEADS + h)
    const int h = bid & (NHEADS - 1);
    const int c = (bid >> 5) & (NCHUNKS - 1);
    const int b = bid >> 10;

    const size_t row0 = (size_t)b * SEQLEN + (size_t)c * CHUNK;
    const float* Bp  = Bmat + row0 * (NGROUPS * DSTATE);            // g = 0
    const float* xp  = x    + row0 * (NHEADS * HEADDIM) + (size_t)h * HEADDIM;
    const size_t dtoff = (((size_t)b * NHEADS + h) * NCHUNKS + c) * CHUNK;
    const float* dtp = dt + dtoff;
    const float* dAp = dA + dtoff;

    // scale[l] = exp(dA_last - dA[l]) * dt[l]   (one l per thread)
    {
        const float dlast = dAp[CHUNK - 1];
        sScale[tid] = __expf(dlast - dAp[tid]) * dtp[tid];
    }
    __syncthreads();

    const int lane  = tid & 31;
    const int wave  = tid >> 5;
    const int hf    = lane >> 4;          // lane half: selects K pair / M group
    const int m     = lane & 15;
    // 2x2 tiles per wave: 2 wave-rows (p) x 4 wave-cols (n)
    const int pbase = (wave >> 2) << 5;   // 0 or 32; tiles at pbase, pbase+16
    const int nbase = (wave & 3) << 4;    // 0..48;   tiles at nbase, nbase+64

    // wave-uniform scalar condition -> guarantees s_cbranch (tensor ops ignore EXEC)
    const bool isWave0 = (__builtin_amdgcn_readfirstlane(wave) == 0);
    const unsigned ldsB = (unsigned)(uintptr_t)&sB[0];

    v8f acc00 = {}, acc01 = {}, acc10 = {}, acc11 = {};

    for (int hl = 0; hl < 2; ++hl) {
        const int lbase = hl * LHALF;

        // ---- TDM: DMA B tile (128 x 128 f32) global -> LDS with HW padding ----
        if (isWave0) {
            const unsigned long long ga =
                (unsigned long long)(uintptr_t)(Bp + (size_t)lbase * (NGROUPS * DSTATE));
            u32x4 g0 = { 1u,                                   // count=1 (valid), user mode
                         ldsB,                                 // lds_addr
                         (unsigned)ga,                         // global_addr[31:0]
                         ((unsigned)(ga >> 32) & 0x01FFFFFFu)  // global_addr[56:32]
                         | (2u << 30) };                       // type=2 (image)
            i32x8 g1 = { (2 << 16)      // data_size = 4B
                       | (1 << 20)      // pad_enable
                       | (6 << 22)      // pad_interval: every 128 dwords
                       | (7 << 25),     // pad_amount: 8 dwords -> stride 136
                         (DSTATE & 0xFFFF) << 16,   // tensor_dim0 = 128
                         (LHALF  & 0xFFFF) << 16,   // tensor_dim1 = 128
                         (DSTATE & 0xFFFF) << 16,   // tile_dim0   = 128
                         LHALF,                     // tile_dim1   = 128
                         DSTATE,                    // tensor_dim0_stride = 128
                         0, 0 };
            i32x4 gz = { 0, 0, 0, 0 };
#if defined(__clang_major__) && __clang_major__ >= 23
            i32x8 gz8 = { 0, 0, 0, 0, 0, 0, 0, 0 };
            __builtin_amdgcn_tensor_load_to_lds(g0, g1, gz, gz, gz8, 0);
#else
            __builtin_amdgcn_tensor_load_to_lds(g0, g1, gz, gz, 0);
#endif
        }

        // ---- stage A = x * scale (128 x 64 f32), non-temporal reads ----
        #pragma unroll
        for (int i = 0; i < 8; ++i) {
            const int idx = tid + 256 * i;       // 0..2047 float4 slots
            const int row = idx >> 4;
            const int c4  = (idx & 15) << 2;
            v4f v = __builtin_nontemporal_load(
                (const v4f*)(xp + (size_t)(lbase + row) * (NHEADS * HEADDIM) + c4));
            v *= sScale[lbase + row];
            *(v4f*)(&sA[row * SA_STRIDE + c4]) = v;
        }

        if (isWave0) __builtin_amdgcn_s_wait_tensorcnt(0);
        __syncthreads();

        // ---- 32 K-steps of V_WMMA_F32_16X16X4_F32 over this l-half ----
        #pragma unroll 4
        for (int k = 0; k < LHALF / 4; ++k) {
            const int ra = k * 4 + 2 * hf;       // K row base for this lane half
            // A frags: ds_load_2addr-friendly {row, row+1} pairs
            const float* ap = &sA[ra * SA_STRIDE + pbase + m];
            v2f a0, a1;
            a0[0] = ap[0];    a0[1] = ap[SA_STRIDE];
            a1[0] = ap[16];   a1[1] = ap[16 + SA_STRIDE];
            // B frags: same {row, row+1} pairing (offset1 = 136 dwords < 256)
            const float* bp = &sB[ra * SB_STRIDE + nbase + m];
            v2f b0, b1;
            b0[0] = bp[0];    b0[1] = bp[SB_STRIDE];
            b1[0] = bp[64];   b1[1] = bp[64 + SB_STRIDE];

            acc00 = __builtin_amdgcn_wmma_f32_16x16x4_f32(false, a0, false, b0, (short)0, acc00, false, false);
            acc01 = __builtin_amdgcn_wmma_f32_16x16x4_f32(false, a0, false, b1, (short)0, acc01, false, false);
            acc10 = __builtin_amdgcn_wmma_f32_16x16x4_f32(false, a1, false, b0, (short)0, acc10, false, false);
            acc11 = __builtin_amdgcn_wmma_f32_16x16x4_f32(false, a1, false, b1, (short)0, acc11, false, false);
        }
        __syncthreads();   // protect sA/sB before next stage overwrites
    }

    // ---- write out D tiles: out[b,c,h,p,n], non-temporal (written once) ----
    const size_t obase = (size_t)bid * (HEADDIM * DSTATE);
    #pragma unroll
    for (int v = 0; v < 8; ++v) {
        const int p0 = pbase + v + 8 * hf;       // first p-tile row
        const int n0 = nbase + m;
        float* op0 = out + obase + (size_t)p0 * DSTATE + n0;
        float* op1 = op0 + 16 * DSTATE;          // second p-tile row (p0+16)
        __builtin_nontemporal_store(acc00[v], op0);
        __builtin_nontemporal_store(acc01[v], op0 + 64);
        __builtin_nontemporal_store(acc10[v], op1);
        __builtin_nontemporal_store(acc11[v], op1 + 64);
    }
}

extern "C" void kernel_launch(void* const* d_in, const int* in_sizes, int n_in,
                              void* d_out, int out_size, void* d_ws, size_t ws_size,
                              hipStream_t stream) {
    const float* B  = (const float*)d_in[0];
    const float* x  = (const float*)d_in[1];
    const float* dt = (const float*)d_in[2];
    const float* dA = (const float*)d_in[3];
    float* out = (float*)d_out;

    dim3 grid(BATCH * NCHUNKS * NHEADS);   // 4096 blocks, one (b,c,h) each
    dim3 block(256);                       // 8 waves (wave32)
    mamba_chunk_state_wmma<<<grid, block, 0, stream>>>(B, x, dt, dA, out);
}